// S2Conv_33036888441086
// MI455X (gfx1250) — compile-verified
//
#include <hip/hip_runtime.h>
#include <hip/hip_bf16.h>

typedef __attribute__((ext_vector_type(2))) float v2f;
typedef __attribute__((ext_vector_type(4))) float v4f;
typedef __attribute__((ext_vector_type(8))) float v8f;

#define B_DIM 128
#define FIN   512
#define FOUT  512
#define NGRID 100
#define S2DIM 64
#define OUTC  680
#define BF    (B_DIM * FIN)   // 65536
#define FG    (FIN * FOUT)    // 262144

// ---- CDNA5 async global->LDS path (guarded: toolchains differ) -------------
#if defined(__has_builtin)
#  if __has_builtin(__builtin_amdgcn_global_load_async_to_lds_b32) && \
      __has_builtin(__builtin_amdgcn_s_wait_asynccnt)
#    define HAVE_ASYNC_LDS 1
#  endif
#endif
#ifndef HAVE_ASYNC_LDS
#  define HAVE_ASYNC_LDS 0
#endif

#if HAVE_ASYNC_LDS
typedef __attribute__((address_space(1))) int* glb_i32p;
typedef __attribute__((address_space(3))) int* lds_i32p;
__device__ __forceinline__ void async_cp_b32(const float* g, float* l) {
    __builtin_amdgcn_global_load_async_to_lds_b32(
        (glb_i32p)g, (lds_i32p)l, /*imm offset*/ 0, /*cpol*/ 0);
}
#endif

// ---------------------------------------------------------------------------
// Kernel 0: xT[i][b][f] = x[b][f][i]   (so K-dim f is contiguous for staging)
// ---------------------------------------------------------------------------
__global__ void k_transpose_x(const float* __restrict__ x, float* __restrict__ xT) {
    __shared__ float t[64][65];
    const int bf0 = blockIdx.x * 64;
    const int tid = threadIdx.x;
    {
        const int col = tid & 63;   // i
        const int r0  = tid >> 6;   // 0..3
#pragma unroll
        for (int p = 0; p < 16; ++p) {
            const int row = r0 + 4 * p;
            t[row][col] = x[(size_t)(bf0 + row) * S2DIM + col];
        }
    }
    __syncthreads();
    {
        const int rowp = tid & 63;  // bf local
        const int c0   = tid >> 6;
#pragma unroll
        for (int p = 0; p < 16; ++p) {
            const int colp = c0 + 4 * p;   // i
            xT[(size_t)colp * BF + bf0 + rowp] = t[rowp][colp];
        }
    }
}

// ---------------------------------------------------------------------------
// Kernel 1: psiT[i][f*512+g] = 0.1 * sum_n w[f,g,n] * Y[n,i]
// GEMM M=262144 (fg rows, w-contiguous), N=64 (i), K=100; 25 WMMA steps.
// ---------------------------------------------------------------------------
__global__ void k_psi(const float* __restrict__ w, const float* __restrict__ Y,
                      float* __restrict__ psiT) {
    __shared__ float Yt[64 * 102];
    const int tid = threadIdx.x;
    for (int idx = tid; idx < NGRID * S2DIM; idx += 256) {
        const int n = idx >> 6;
        const int i = idx & 63;
        Yt[i * 102 + n] = Y[idx];
    }
    __syncthreads();

    const int lane = tid & 31;
    const int wv   = tid >> 5;                    // wave32
    const int row0 = blockIdx.x * 128 + wv * 16;  // fg-row tile base
    const int i0   = blockIdx.y * 16;             // i tile base
    const int arow = row0 + (lane & 15);
    const int brow = (i0 + (lane & 15)) * 102;
    const int koff = (lane < 16) ? 0 : 2;         // frag K split at lane 16

    v8f c = {};
#pragma unroll
    for (int kk = 0; kk < 25; ++kk) {
        const int k0 = kk * 4 + koff;
        v2f a = *(const v2f*)(w + (size_t)arow * NGRID + k0);
        v2f b = *(const v2f*)(&Yt[brow + k0]);
        c = __builtin_amdgcn_wmma_f32_16x16x4_f32(false, a, false, b, (short)0,
                                                  c, false, false);
    }

    const float s  = 0.1f;  // 1/sqrt(100)
    const int   ig = i0 + (lane & 15);
    const int   rb = row0 + ((lane < 16) ? 0 : 8);
    float* dst = psiT + (size_t)ig * FG + rb;
    v4f lo = {c[0] * s, c[1] * s, c[2] * s, c[3] * s};
    v4f hi = {c[4] * s, c[5] * s, c[6] * s, c[7] * s};
    *(v4f*)(dst)     = lo;   // rows rb..rb+3
    *(v4f*)(dst + 4) = hi;   // rows rb+4..rb+7
}

// ---------------------------------------------------------------------------
// Kernel 2 (dominant): per l, per (u,m):
//   o[g,b, off+u*d+m] = (1/sqrt(512)) * sum_f xT[s+u][b][f] * psiT[s+m][f*512+g]
// 64(b)x64(g) macro-tile per workgroup; K in 32-chunks, double-buffered LDS
// filled by async global->LDS copies (ASYNCcnt), overlapped with WMMA.
// ---------------------------------------------------------------------------
__global__ void k_s2conv_main(const float* __restrict__ xT,
                              const float* __restrict__ psiT,
                              float* __restrict__ out,
                              int s, int d, int off) {
    __shared__ float Asm[2][64 * 34];  // [buf][b_local][k_local]
    __shared__ float Bsm[2][64 * 34];  // [buf][g_local][k_local]

    const int tid  = threadIdx.x;
    const int lane = tid & 31;
    const int wv   = tid >> 5;

    const int tile = blockIdx.x;           // 0..15
    const int b0   = (tile & 1) * 64;
    const int g0   = (tile >> 1) * 64;
    const int um   = blockIdx.y;
    const int u    = um / d;
    const int m    = um - u * d;

    const float* xcol = xT   + (size_t)(s + u) * BF + (size_t)b0 * FIN;  // [b][f]
    const float* pcol = psiT + (size_t)(s + m) * FG + g0;                // [f][g]

    // staging mapping: 8 elements per thread for each of A and B
    const int klA = tid & 31, blA = tid >> 5;   // A[blA+8p][klA], coalesced over f
    const int glB = tid & 63, klB = tid >> 6;   // B[glB][klB+4p], coalesced over g

    const int sg   = wv & 3;              // g sub-tile 0..3
    const int sb   = (wv >> 2) * 2;       // b sub-tiles {sb, sb+1}
    const int koff = (lane < 16) ? 0 : 2;
    const int rA0  = (sb * 16 + (lane & 15)) * 34;
    const int rA1  = rA0 + 16 * 34;
    const int rB   = (sg * 16 + (lane & 15)) * 34;

    v8f c0 = {};
    v8f c1 = {};

    auto stage = [&](int buf, int f0) {
        const float* sa = xcol + (size_t)blA * FIN + f0 + klA;
        const float* sp = pcol + (size_t)(f0 + klB) * FOUT + glB;
        float* da = &Asm[buf][blA * 34 + klA];
        float* db = &Bsm[buf][glB * 34 + klB];
#pragma unroll
        for (int p = 0; p < 8; ++p) {
#if HAVE_ASYNC_LDS
            async_cp_b32(sa + (size_t)(8 * p) * FIN, da + (8 * p) * 34);
            async_cp_b32(sp + (size_t)(4 * p) * FOUT, db + 4 * p);
#else
            da[(8 * p) * 34] = sa[(size_t)(8 * p) * FIN];
            db[4 * p]        = sp[(size_t)(4 * p) * FOUT];
#endif
        }
    };

    stage(0, 0);
    for (int c = 0; c < 16; ++c) {
        const int cur = c & 1;
#if HAVE_ASYNC_LDS
        __builtin_amdgcn_s_wait_asynccnt(0);  // own chunk-c copies landed in LDS
#endif
        __syncthreads();                      // all waves' chunk-c data visible
        if (c + 1 < 16) stage(cur ^ 1, (c + 1) * 32);  // overlap with compute

        const float* A  = Asm[cur];
        const float* Bm = Bsm[cur];
#pragma unroll
        for (int kk = 0; kk < 8; ++kk) {
            const int k0 = kk * 4 + koff;
            v2f a0 = *(const v2f*)(&A[rA0 + k0]);
            v2f a1 = *(const v2f*)(&A[rA1 + k0]);
            v2f b  = *(const v2f*)(&Bm[rB + k0]);
            c0 = __builtin_amdgcn_wmma_f32_16x16x4_f32(false, a0, false, b,
                                                       (short)0, c0, false, false);
            c1 = __builtin_amdgcn_wmma_f32_16x16x4_f32(false, a1, false, b,
                                                       (short)0, c1, false, false);
        }
    }

    const float scale = 0.04419417382415922f;  // 1/sqrt(512)
    const int gg   = g0 + sg * 16 + (lane & 15);            // N dim = g
    const int chan = off + u * d + m;
    const int bb0  = b0 + sb * 16 + ((lane < 16) ? 0 : 8);  // M dim = b
    float* o0 = out + (size_t)gg * (B_DIM * OUTC) + (size_t)bb0 * OUTC + chan;
    float* o1 = o0 + (size_t)16 * OUTC;
#pragma unroll
    for (int r = 0; r < 8; ++r) {
        o0[(size_t)r * OUTC] = c0[r] * scale;
        o1[(size_t)r * OUTC] = c1[r] * scale;
    }
}

// ---------------------------------------------------------------------------
extern "C" void kernel_launch(void* const* d_in, const int* in_sizes, int n_in,
                              void* d_out, int out_size, void* d_ws, size_t ws_size,
                              hipStream_t stream) {
    const float* x = (const float*)d_in[0];  // (128, 512, 64)
    const float* w = (const float*)d_in[1];  // (512, 512, 100)
    const float* Y = (const float*)d_in[2];  // (100, 64)
    float* out = (float*)d_out;              // (512, 128, 680)

    float* psiT = (float*)d_ws;              // 64*512*512 floats = 64 MB
    float* xT   = psiT + (size_t)S2DIM * FG; // 64*128*512 floats = 16 MB

    // Stage 0: transpose x -> xT[i][b][f]
    k_transpose_x<<<BF / 64, 256, 0, stream>>>(x, xT);

    // Stage 1: psiT[i][f*512+g]
    k_psi<<<dim3(FG / 128, S2DIM / 16), 256, 0, stream>>>(w, Y, psiT);

    // Stage 2: main contraction, one launch per l
    static const int off[8] = {0, 1, 10, 35, 84, 165, 286, 455};
    for (int l = 0; l <= 7; ++l) {
        const int d = 2 * l + 1;
        k_s2conv_main<<<dim3(16, d * d), 256, 0, stream>>>(xT, psiT, out,
                                                           l * l, d, off[l]);
    }
}